// Model_60309930771013
// MI455X (gfx1250) — compile-verified
//
#include <hip/hip_runtime.h>
#include <hip/hip_bf16.h>

// 5x5 box filter, reflect padding, 8192x8192 fp32.
// Memory-bound: 536 MB HBM traffic -> ~23 us floor @ 23.3 TB/s.
// Strategy: async global->LDS halo staging (gfx1250 ASYNCcnt path),
//   b128 row staging for interior tiles, reflect b32 gather for border tiles,
//   separable sliding-window sums in registers, nontemporal b128 stores.

namespace {
constexpr int kH = 8192;
constexpr int kW = 8192;
constexpr int TX = 128;          // output tile width
constexpr int TY = 32;           // output tile height
constexpr int LC = TX + 4;       // 132 LDS columns (halo)
constexpr int LR = TY + 4;       // 36 LDS rows (halo)
constexpr int LDSN = LC * LR;    // 4752 floats (~19 KB)
constexpr int CHUNKS = LC / 4;   // 33 b128 chunks per LDS row
constexpr int NTHREADS = 256;
}

typedef float v4f __attribute__((ext_vector_type(4)));
typedef int   v4i __attribute__((ext_vector_type(4)));
typedef __attribute__((address_space(1))) int* gptr_i32;
typedef __attribute__((address_space(3))) int* lptr_i32;
typedef __attribute__((address_space(1))) v4i* gptr_v4i;
typedef __attribute__((address_space(3))) v4i* lptr_v4i;

#if defined(__AMDGCN__) && __has_builtin(__builtin_amdgcn_global_load_async_to_lds_b32)
#define USE_ASYNC_LDS 1
#else
#define USE_ASYNC_LDS 0
#endif

#if defined(__AMDGCN__) && __has_builtin(__builtin_amdgcn_global_load_async_to_lds_b128)
#define USE_ASYNC_LDS_B128 1
#else
#define USE_ASYNC_LDS_B128 0
#endif

__device__ __forceinline__ int reflect_idx(int i, int n) {
  // radius (2) < n, so a single reflection suffices; matches jnp.pad 'reflect'
  i = (i < 0) ? -i : i;
  return (i >= n) ? (2 * n - 2 - i) : i;
}

__global__ __launch_bounds__(NTHREADS) void box5x5_kernel(
    const float* __restrict__ in, float* __restrict__ out) {
  __shared__ __align__(16) float tile[LDSN];

  const int tid  = threadIdx.x;
  const int col0 = blockIdx.x * TX;
  const int row0 = blockIdx.y * TY;

  // Tile is "interior" when the whole halo lies in-bounds (no reflection).
  const bool interior = (row0 >= 2) && (row0 + TY + 2 <= kH) &&
                        (col0 >= 2) && (col0 + TX + 2 <= kW);

#if USE_ASYNC_LDS_B128
  if (interior) {
    // ---- Fast path: stage 36 contiguous 132-float rows as b128 asyncs ----
    // Global row start is 8B-aligned (legal for b128); LDS dest is 16B-aligned.
    const float* rowbase = in + (size_t)(row0 - 2) * kW + (col0 - 2);
    for (int e = tid; e < LR * CHUNKS; e += NTHREADS) {
      const int r = e / CHUNKS;
      const int k = e - r * CHUNKS;
      const float* src = rowbase + (size_t)r * kW + (k << 2);
      __builtin_amdgcn_global_load_async_to_lds_b128(
          (gptr_v4i)src,
          (lptr_v4i)&tile[r * LC + (k << 2)],
          /*imm offset*/ 0, /*cpol*/ 0);
    }
  } else
#endif
  {
    // ---- Generic path: per-element reflect gather (border tiles) ----
    for (int e = tid; e < LDSN; e += NTHREADS) {
      const int r = e / LC;
      const int c = e - r * LC;
      const int gy = reflect_idx(row0 - 2 + r, kH);
      const int gx = reflect_idx(col0 - 2 + c, kW);
      const float* src = in + (size_t)gy * kW + gx;
#if USE_ASYNC_LDS
      __builtin_amdgcn_global_load_async_to_lds_b32(
          (gptr_i32)src,
          (lptr_i32)&tile[e],
          /*imm offset*/ 0, /*cpol*/ 0);
#else
      tile[e] = __ldg(src);
#endif
    }
  }

#if USE_ASYNC_LDS || USE_ASYNC_LDS_B128
#if __has_builtin(__builtin_amdgcn_s_wait_asynccnt)
  __builtin_amdgcn_s_wait_asynccnt(0);
#else
  asm volatile("s_wait_asynccnt 0" ::: "memory");
#endif
#endif
  __syncthreads();

  // ---- Each thread: 4x4 output micro-tile ----
  const int cg = tid & 31;   // 32 column groups of 4
  const int rg = tid >> 5;   // 8 row groups of 4
  const int x0 = cg << 2;    // tile-local output col base
  const int y0 = rg << 2;    // tile-local output row base

  // Horizontal 5-tap sums for 8 halo rows (y0 .. y0+7 in LDS coords),
  // 4 adjacent output columns, via two aligned 16B LDS reads per row.
  float h[8][4];
#pragma unroll
  for (int j = 0; j < 8; ++j) {
    const v4f* p = (const v4f*)&tile[(y0 + j) * LC + x0];  // 16B aligned
    const v4f a = p[0];
    const v4f b = p[1];
    const float h0 = a.x + a.y + a.z + a.w + b.x;
    h[j][0] = h0;
    h[j][1] = h0 - a.x + b.y;
    h[j][2] = h[j][1] - a.y + b.z;
    h[j][3] = h[j][2] - a.z + b.w;
  }

  // Vertical 5-tap sliding sum, scale by 1/25.
  constexpr float inv25 = 1.0f / 25.0f;
  v4f o[4];
#pragma unroll
  for (int c = 0; c < 4; ++c) {
    float v = h[0][c] + h[1][c] + h[2][c] + h[3][c] + h[4][c];
    o[0][c] = v * inv25;
    v += h[5][c] - h[0][c]; o[1][c] = v * inv25;
    v += h[6][c] - h[1][c]; o[2][c] = v * inv25;
    v += h[7][c] - h[2][c]; o[3][c] = v * inv25;
  }

  // Write-once output: nontemporal 16B stores (don't rinse the 192MB L2).
  float* obase = out + (size_t)(row0 + y0) * kW + (col0 + x0);
#pragma unroll
  for (int i = 0; i < 4; ++i) {
    __builtin_nontemporal_store(o[i], (v4f*)(obase + (size_t)i * kW));
  }
}

extern "C" void kernel_launch(void* const* d_in, const int* in_sizes, int n_in,
                              void* d_out, int out_size, void* d_ws, size_t ws_size,
                              hipStream_t stream) {
  (void)in_sizes; (void)n_in; (void)out_size; (void)d_ws; (void)ws_size;
  const float* in = (const float*)d_in[0];
  float* out = (float*)d_out;
  dim3 grid(kW / TX, kH / TY);  // 64 x 256 tiles
  box5x5_kernel<<<grid, dim3(NTHREADS), 0, stream>>>(in, out);
}